// AtomMpnn_91156385890475
// MI455X (gfx1250) — compile-verified
//
#include <hip/hip_runtime.h>
#include <hip/hip_bf16.h>
#include <math.h>

typedef __attribute__((ext_vector_type(2))) float v2f;
typedef __attribute__((ext_vector_type(8))) float v8f;

#define NDIM 4
#define MDIM 64
#define ADIM 64
#define ODIM 5
#define FDIM 256
#define AO   320          // ADIM*ODIM = contraction / row dim
#define GCOL 1024         // MDIM*16  (i padded 9->16)

// ---------------------------------------------------------------------------
// Kernel 1: Sc5[n][a*5+o][b*5+p]  (5x bilinear w*B*w^T + swish, filter, expand)
// ---------------------------------------------------------------------------
__global__ void sc5_kernel(const float* __restrict__ S,
                           const float* __restrict__ wst,
                           float* __restrict__ Sc5) {
    int t = blockIdx.x * blockDim.x + threadIdx.x;
    if (t >= NDIM * ADIM * ADIM) return;
    int n  = t / (ADIM * ADIM);
    int ai = (t / ADIM) % ADIM;
    int aj = t % ADIM;

    float B[3][3];
#pragma unroll
    for (int r = 0; r < 3; ++r)
#pragma unroll
        for (int c = 0; c < 3; ++c)
            B[r][c] = S[((((size_t)n * ADIM + ai) * 3 + r) * ADIM + aj) * 3 + c];

#pragma unroll
    for (int it = 0; it < 5; ++it) {
        const float* w = wst + it * 9;
        float T[3][3];
#pragma unroll
        for (int a = 0; a < 3; ++a)
#pragma unroll
            for (int d = 0; d < 3; ++d)
                T[a][d] = w[a * 3 + 0] * B[0][d] + w[a * 3 + 1] * B[1][d] + w[a * 3 + 2] * B[2][d];
#pragma unroll
        for (int a = 0; a < 3; ++a)
#pragma unroll
            for (int c = 0; c < 3; ++c) {
                float v = T[a][0] * w[c * 3 + 0] + T[a][1] * w[c * 3 + 1] + T[a][2] * w[c * 3 + 2];
                B[a][c] = v / (1.0f + __expf(-v));   // swish
            }
    }

    const int idx5[5] = {0, 1, 2, 2, 2};
#pragma unroll
    for (int o = 0; o < 5; ++o) {
        int ro = idx5[o];
#pragma unroll
        for (int p = 0; p < 5; ++p) {
            int cp = idx5[p];
            float flt = ((ro < 2 && cp < 2) || (ro == 2 && cp == 2)) ? 1.0f : 0.0f;
            Sc5[((size_t)n * AO + ai * 5 + o) * AO + (aj * 5 + p)] = flt * B[ro][cp];
        }
    }
}

// ---------------------------------------------------------------------------
// Kernel 2: coup[(n,m,b)][k] = Y_k(R) * ||C[n,m,b,:]||
// ---------------------------------------------------------------------------
__global__ void coup_kernel(const float* __restrict__ C,
                            const float* __restrict__ R,
                            float* __restrict__ coup) {
    int t = blockIdx.x * blockDim.x + threadIdx.x;
    if (t >= NDIM * MDIM * ADIM) return;

    const float* Cp = C + (size_t)t * ODIM;
    float s = 0.f;
#pragma unroll
    for (int o = 0; o < ODIM; ++o) s += Cp[o] * Cp[o];
    float cn = sqrtf(s);

    const float* Rp = R + (size_t)t * 3;
    float rx = Rp[0], ry = Rp[1], rz = Rp[2];
    float rn = sqrtf(rx * rx + ry * ry + rz * rz) + 1e-12f;
    float x = rx / rn, y = ry / rn, z = rz / rn;

    const float c1 = 0.4886025119029199f;
    const float c2 = 1.0925484305920792f;
    float Yv[9];
    Yv[0] = 0.28209479177387814f;
    Yv[1] = c1 * y; Yv[2] = c1 * z; Yv[3] = c1 * x;
    Yv[4] = c2 * x * y; Yv[5] = c2 * y * z;
    Yv[6] = 0.31539156525252005f * (3.0f * z * z - 1.0f);
    Yv[7] = c2 * x * z;
    Yv[8] = 0.5462742152960396f * (x * x - y * y);

#pragma unroll
    for (int k = 0; k < 9; ++k) coup[(size_t)t * 9 + k] = Yv[k] * cn;
}

// ---------------------------------------------------------------------------
// Kernel 3: G'[n][b*5+p][m*16+i] = sum_k coup[n,m,b,k]*cgc[k,i,segid[p]] (i<9)
// ---------------------------------------------------------------------------
__global__ void gprime_kernel(const float* __restrict__ coup,
                              const float* __restrict__ cgc,
                              float* __restrict__ Gp) {
    int idx = blockIdx.x * blockDim.x + threadIdx.x;
    if (idx >= NDIM * AO * GCOL) return;
    int n   = idx / (AO * GCOL);
    int r   = (idx >> 10) % AO;     // b*5+p
    int col = idx & (GCOL - 1);     // m*16+i
    int b = r / 5, p = r % 5;
    int m = col >> 4, i = col & 15;

    float v = 0.f;
    if (i < 9) {
        const int segid[5] = {0, 0, 1, 2, 3};
        int q = segid[p];
        const float* cp = coup + (((size_t)n * MDIM + m) * ADIM + b) * 9;
#pragma unroll
        for (int k = 0; k < 9; ++k) v += cp[k] * cgc[((size_t)k * 9 + i) * 9 + q];
    }
    Gp[idx] = v;   // Gp flat layout == thread index
}

// ---------------------------------------------------------------------------
// Kernel 4 (WMMA): Keff[n][ao][m][i16] = C[n,m,ao] * sum_bp Sc5[n,ao,bp]*G'[n,bp,m*16+i]
// one wave per (n, ao-tile of 16, m); K = 320 in steps of 4 (f32 WMMA)
// ---------------------------------------------------------------------------
__global__ void phaseC_kernel(const float* __restrict__ Sc5,
                              const float* __restrict__ Gp,
                              const float* __restrict__ C,
                              float* __restrict__ Keff) {
    int lane = threadIdx.x & 31;
    int wv = (blockIdx.x * blockDim.x + threadIdx.x) >> 5;
    if (wv >= NDIM * 20 * MDIM) return;
    int n   = wv / (20 * MDIM);
    int rem = wv % (20 * MDIM);
    int aoT = rem / MDIM;
    int m   = rem % MDIM;
    int l15  = lane & 15;
    int half = lane >> 4;

    // A: Sc5 row = aoT*16 + (lane&15); cols = k0 + 2*half {,+1} -> contiguous float2
    const float* Arow  = Sc5 + ((size_t)n * AO + aoT * 16 + l15) * AO;
    // B: G' row = k; col = m*16 + (lane&15)
    const float* Bbase = Gp + (size_t)n * AO * GCOL + (size_t)m * 16 + l15;

    v8f acc = {0.f, 0.f, 0.f, 0.f, 0.f, 0.f, 0.f, 0.f};
#pragma unroll 4
    for (int k0 = 0; k0 < AO; k0 += 4) {
        int ka = k0 + 2 * half;
        v2f a = *(const v2f*)(Arow + ka);
        v2f b;
        b.x = Bbase[(size_t)ka * GCOL];
        b.y = Bbase[(size_t)(ka + 1) * GCOL];
        acc = __builtin_amdgcn_wmma_f32_16x16x4_f32(false, a, false, b, (short)0, acc,
                                                    false, false);
    }

    // Epilogue: scale by C[n,m,ao] (row = ao), store A-matrix-friendly layout
    const float* Crow = C + ((size_t)n * MDIM + m) * AO;
#pragma unroll
    for (int j = 0; j < 8; ++j) {
        int row = aoT * 16 + j + 8 * half;                 // D row -> ao
        float v = acc[j] * Crow[row];
        Keff[(((size_t)n * AO + row) * MDIM + m) * 16 + l15] = v;  // col -> i
    }
}

// ---------------------------------------------------------------------------
// Kernel 5 (WMMA): out[n][m][f][i(9)] = sum_ao Keff[n,ao,m,i] * ao_emb[n,m,ao,f]
// one wave per (n, m, pair of f-tiles): A loaded once per k-step, used by 2 WMMAs.
// Streams the 80 MB tensor exactly once with non-temporal loads.
// ---------------------------------------------------------------------------
__global__ void phaseD_kernel(const float* __restrict__ Keff,
                              const float* __restrict__ aoe,
                              float* __restrict__ out) {
    int lane = threadIdx.x & 31;
    int wv = (blockIdx.x * blockDim.x + threadIdx.x) >> 5;
    if (wv >= NDIM * MDIM * 8) return;   // 2048 waves, 2 f-tiles each
    int n   = wv >> 9;                   // / (64*8)
    int rem = wv & 511;
    int m   = rem >> 3;
    int fG  = rem & 7;                   // f-tile pair: f = fG*32 .. fG*32+31
    int l15  = lane & 15;
    int half = lane >> 4;

    // A: Keff[n][k][m][i=lane&15], k varies -> stride 1024 floats (two 64B segs/load)
    const float* Abase = Keff + ((size_t)n * AO * MDIM + (size_t)m) * 16 + l15;
    // B: ao_emb[n][m][k][f], k stride 256 floats; two adjacent 16-wide f tiles
    const float* Bbase = aoe + (size_t)(n * MDIM + m) * (AO * FDIM) + fG * 32 + l15;

    v8f acc0 = {0.f, 0.f, 0.f, 0.f, 0.f, 0.f, 0.f, 0.f};
    v8f acc1 = {0.f, 0.f, 0.f, 0.f, 0.f, 0.f, 0.f, 0.f};
#pragma unroll 8
    for (int k0 = 0; k0 < AO; k0 += 4) {
        int ka = k0 + 2 * half;
        v2f a;
        a.x = Abase[(size_t)ka * (MDIM * 16)];
        a.y = Abase[(size_t)(ka + 1) * (MDIM * 16)];
        const float* Bk0 = Bbase + (size_t)ka * FDIM;
        const float* Bk1 = Bbase + (size_t)(ka + 1) * FDIM;
        v2f b0, b1;
        b0.x = __builtin_nontemporal_load(Bk0);          // f-tile 0, K=ka
        b0.y = __builtin_nontemporal_load(Bk1);          // f-tile 0, K=ka+1
        b1.x = __builtin_nontemporal_load(Bk0 + 16);     // f-tile 1, K=ka
        b1.y = __builtin_nontemporal_load(Bk1 + 16);     // f-tile 1, K=ka+1
        acc0 = __builtin_amdgcn_wmma_f32_16x16x4_f32(false, a, false, b0, (short)0, acc0,
                                                     false, false);
        acc1 = __builtin_amdgcn_wmma_f32_16x16x4_f32(false, a, false, b1, (short)0, acc1,
                                                     false, false);
    }

    int f0 = fG * 32 + l15;
    float* orow0 = out + ((size_t)(n * MDIM + m) * FDIM + f0) * 9;
    float* orow1 = orow0 + 16 * 9;       // second f-tile, f0+16
#pragma unroll
    for (int j = 0; j < 8; ++j) {
        int i = j + 8 * half;            // D row -> output channel i
        if (i < 9) {
            __builtin_nontemporal_store(acc0[j], orow0 + i);
            __builtin_nontemporal_store(acc1[j], orow1 + i);
        }
    }
}

// ---------------------------------------------------------------------------
extern "C" void kernel_launch(void* const* d_in, const int* in_sizes, int n_in,
                              void* d_out, int out_size, void* d_ws, size_t ws_size,
                              hipStream_t stream) {
    const float* aoe = (const float*)d_in[0];   // [4,64,64,5,256]
    const float* C   = (const float*)d_in[1];   // [4,64,64,5]
    const float* S   = (const float*)d_in[2];   // [4,64,3,64,3]
    const float* R   = (const float*)d_in[3];   // [4,64,64,3]
    const float* wst = (const float*)d_in[4];   // [5,3,3]
    const float* cgc = (const float*)d_in[5];   // [9,9,9]
    float* out = (float*)d_out;                 // [4,64,256,9]

    char* ws = (char*)d_ws;
    float* Sc5  = (float*)(ws + 0);         // 4*320*320*4        = 1,638,400 B
    float* coup = (float*)(ws + 1638400);   // 16384*9*4          =   589,824 B
    float* Gp   = (float*)(ws + 2228224);   // 4*320*1024*4       = 5,242,880 B
    float* Keff = (float*)(ws + 7471104);   // 4*320*64*16*4      = 5,242,880 B
                                            // total ~12.7 MB of d_ws

    sc5_kernel   <<<  64, 256, 0, stream>>>(S, wst, Sc5);           // 16384 threads
    coup_kernel  <<<  64, 256, 0, stream>>>(C, R, coup);            // 16384 threads
    gprime_kernel<<<5120, 256, 0, stream>>>(coup, cgc, Gp);         // 1,310,720 threads
    phaseC_kernel<<< 640, 256, 0, stream>>>(Sc5, Gp, C, Keff);      // 5120 waves
    phaseD_kernel<<< 256, 256, 0, stream>>>(Keff, aoe, out);        // 2048 waves x 2 tiles
}